// LRUCell_33861522161927
// MI455X (gfx1250) — compile-verified
//
#include <hip/hip_runtime.h>
#include <math.h>

// ---------------------------------------------------------------------------
// LRU cell for MI455X (gfx1250, wave32).
//   Bu = x @ B_norm^T  via V_WMMA_F32_16X16X4_F32 (fp32 WMMA, exact precision)
//   h_t = Lambda * h_{t-1} + Bu_t  via chunked scan (aggregate / combine / emit)
// Bandwidth plan: never materialize Bu (recompute GEMM in both scan passes).
// Traffic ~= 2*x (64MB) + out (128MB) ~ 200MB -> ~8.6us at 23.3 TB/s.
// ---------------------------------------------------------------------------

typedef __attribute__((ext_vector_type(2))) float v2f;
typedef __attribute__((ext_vector_type(8))) float v8f;

#define T_TOTAL 32768
#define D_IN    256
#define D_H     512
#define CHUNK_T 128                  // timesteps per scan chunk
#define NCHUNK  (T_TOTAL / CHUNK_T)  // 256
#define NTILE   (CHUNK_T / 16)       // 8 WMMA time-tiles per chunk
#define LDSB_STRIDE 260              // 256 + 4 pad: avoids 64-bank conflicts

__device__ __forceinline__ float2 cmul(float2 a, float2 b) {
    return make_float2(a.x * b.x - a.y * b.y, a.x * b.y + a.y * b.x);
}

// ---------------------------------------------------------------------------
// K1: per-channel setup.  lambda[c] = exp(-exp(nu)+i*exp(theta)),
//     BnR/BnI[c][k] = B_{r,i}[c][k] * exp(gamma[c])
// ---------------------------------------------------------------------------
__global__ __launch_bounds__(D_IN) void setup_kernel(
    const float* __restrict__ nu_log, const float* __restrict__ theta_log,
    const float* __restrict__ gamma_log,
    const float* __restrict__ B_real, const float* __restrict__ B_img,
    float2* __restrict__ lambda, float* __restrict__ BnR, float* __restrict__ BnI)
{
    const int c = blockIdx.x;
    const int k = threadIdx.x;
    const float nu  = __expf(nu_log[c]);
    const float th  = __expf(theta_log[c]);
    const float mag = __expf(-nu);
    if (k == 0) lambda[c] = make_float2(mag * __cosf(th), mag * __sinf(th));
    const float g = __expf(gamma_log[c]);
    BnR[c * D_IN + k] = B_real[c * D_IN + k] * g;
    BnI[c * D_IN + k] = B_img[c * D_IN + k] * g;
}

// ---------------------------------------------------------------------------
// K2/K4: one wave per block; block = (time chunk) x (16 channels).
// Computes Bu on the fly with fp32 WMMA, scans time in-register.
// WRITE_OUT=false: emit chunk aggregate.  WRITE_OUT=true: seeded, emit h_t.
// ---------------------------------------------------------------------------
template <bool WRITE_OUT>
__global__ __launch_bounds__(32) void chunk_scan_kernel(
    const float*  __restrict__ x,
    const float*  __restrict__ BnR, const float* __restrict__ BnI,
    const float2* __restrict__ lambda,
    const float2* __restrict__ carry,   // [NCHUNK][D_H], only WRITE_OUT
    float2*       __restrict__ agg,     // [NCHUNK][D_H], only !WRITE_OUT
    float*        __restrict__ out)     // 1024 f (last state) + T*H float2
{
    __shared__ float ldsBR[16 * LDSB_STRIDE];
    __shared__ float ldsBI[16 * LDSB_STRIDE];

    const int chunk = blockIdx.x;
    const int n0    = blockIdx.y * 16;      // first channel of tile
    const int lane  = threadIdx.x;          // wave32
    const int cc    = lane & 15;            // channel-in-tile / A-row index
    const int khalf = lane >> 4;            // K pair select per ISA A layout

    // Stage the 16x256 B_norm tiles (both planes) in LDS, reused by all
    // NTILE time-tiles of this chunk.
    for (int i = lane; i < 16 * D_IN; i += 32) {
        const int n = i >> 8;
        const int k = i & (D_IN - 1);
        ldsBR[n * LDSB_STRIDE + k] = BnR[(n0 + n) * D_IN + k];
        ldsBI[n * LDSB_STRIDE + k] = BnI[(n0 + n) * D_IN + k];
    }
    __syncthreads();

    const float2 lam = lambda[n0 + cc];
    float2 l2 = cmul(lam, lam);
    float2 l4 = cmul(l2, l2);
    const float2 lam8 = cmul(l4, l4);       // Lambda^8 (per-lane-half stride)

    // Carry into this chunk (zero-seeded for the aggregate pass).
    float2 H0 = make_float2(0.f, 0.f);
    if (WRITE_OUT) H0 = carry[chunk * D_H + (n0 + cc)];

    const int t0 = chunk * CHUNK_T;
    const float2* x2 = (const float2*)x;

    for (int tile = 0; tile < NTILE; ++tile) {
        const int tBase = t0 + tile * 16;
        const int rowA  = tBase + cc;       // A-matrix row (time) for this lane
        if (tile + 1 < NTILE)               // gfx1250 global_prefetch_b8
            __builtin_prefetch(x + (rowA + 16) * D_IN, 0, 0);

        v8f cR = {};                        // D/C tiles: 16 time x 16 chan f32
        v8f cI = {};
#pragma unroll 8
        for (int k = 0; k < D_IN; k += 4) {
            // A (16x4 f32): lanes 0-15 hold K=0,1; lanes 16-31 hold K=2,3
            const float2 av = x2[(rowA * D_IN + k) / 2 + khalf];
            v2f a; a.x = av.x; a.y = av.y;
            // B (4x16 f32): symmetric layout, channel = n0 + cc per lane
            const float2 br = *(const float2*)&ldsBR[cc * LDSB_STRIDE + k + 2 * khalf];
            const float2 bi = *(const float2*)&ldsBI[cc * LDSB_STRIDE + k + 2 * khalf];
            v2f b; b.x = br.x; b.y = br.y;
            cR = __builtin_amdgcn_wmma_f32_16x16x4_f32(false, a, false, b,
                                                       (short)0, cR, false, false);
            b.x = bi.x; b.y = bi.y;
            cI = __builtin_amdgcn_wmma_f32_16x16x4_f32(false, a, false, b,
                                                       (short)0, cI, false, false);
        }

        // In-tile time scan.  Lane cc holds times tBase+0..7 for channel
        // n0+cc; lane cc+16 holds times tBase+8..15 for the same channel.
        // Phase 1: zero-seeded local scan (only final value needed).
        float2 s = make_float2(0.f, 0.f);
#pragma unroll
        for (int r = 0; r < 8; ++r) {
            s = cmul(lam, s);
            s.x += cR[r]; s.y += cI[r];
        }
        const float sLx = __shfl(s.x, cc, 32);  // lower half's local final
        const float sLy = __shfl(s.y, cc, 32);

        float2 hin;
        if (lane < 16) {
            hin = H0;
        } else {
            const float2 t = cmul(lam8, H0);
            hin = make_float2(t.x + sLx, t.y + sLy);
        }

        // Phase 2: seeded rescan; emit on the write pass.
        float2 h = hin;
#pragma unroll
        for (int r = 0; r < 8; ++r) {
            h = cmul(lam, h);
            h.x += cR[r]; h.y += cI[r];
            if (WRITE_OUT) {
                const int t = tBase + khalf * 8 + r;
                float2* seq = (float2*)(out + 2 * D_H);
                seq[t * D_H + (n0 + cc)] = h;
                if (t == T_TOTAL - 1)           // last-state output
                    ((float2*)out)[n0 + cc] = h;
            }
        }
        // Tile carry = state after tBase+15, held by the upper half.
        H0.x = __shfl(h.x, cc + 16, 32);
        H0.y = __shfl(h.y, cc + 16, 32);
    }

    if (!WRITE_OUT && lane < 16)
        agg[chunk * D_H + (n0 + cc)] = H0;      // chunk aggregate (carry-in 0)
}

// ---------------------------------------------------------------------------
// K3: sequential combine over 256 chunk aggregates, parallel over channels.
// carry[j][c] = Lambda^CHUNK_T * carry[j-1][c] + agg[j-1][c], carry[0]=0.
// ---------------------------------------------------------------------------
__global__ __launch_bounds__(256) void combine_kernel(
    const float2* __restrict__ lambda, const float2* __restrict__ agg,
    float2* __restrict__ carry)
{
    const int c = blockIdx.x * blockDim.x + threadIdx.x;
    if (c >= D_H) return;
    float2 lamL = lambda[c];
#pragma unroll
    for (int i = 0; i < 7; ++i) lamL = cmul(lamL, lamL);   // Lambda^128
    float2 acc = make_float2(0.f, 0.f);
    carry[c] = acc;
    for (int j = 1; j < NCHUNK; ++j) {
        const float2 a = agg[(j - 1) * D_H + c];
        acc = cmul(lamL, acc);
        acc.x += a.x; acc.y += a.y;
        carry[j * D_H + c] = acc;
    }
}

// ---------------------------------------------------------------------------
// Inputs (setup_inputs order): h0, x, nu_log, theta_log, gamma_log, B_real, B_img
// Output: [last_state (512 c64)] ++ [hidden_states (32768x512 c64)], as floats.
// ---------------------------------------------------------------------------
extern "C" void kernel_launch(void* const* d_in, const int* in_sizes, int n_in,
                              void* d_out, int out_size, void* d_ws, size_t ws_size,
                              hipStream_t stream)
{
    const float* x         = (const float*)d_in[1];
    const float* nu_log    = (const float*)d_in[2];
    const float* theta_log = (const float*)d_in[3];
    const float* gamma_log = (const float*)d_in[4];
    const float* B_real    = (const float*)d_in[5];
    const float* B_img     = (const float*)d_in[6];

    char* ws = (char*)d_ws;
    size_t off = 0;
    float2* lambda = (float2*)(ws + off); off += (size_t)D_H * sizeof(float2);
    float*  BnR    = (float*)(ws + off);  off += (size_t)D_H * D_IN * sizeof(float);
    float*  BnI    = (float*)(ws + off);  off += (size_t)D_H * D_IN * sizeof(float);
    float2* agg    = (float2*)(ws + off); off += (size_t)NCHUNK * D_H * sizeof(float2);
    float2* carry  = (float2*)(ws + off); off += (size_t)NCHUNK * D_H * sizeof(float2);

    setup_kernel<<<D_H, D_IN, 0, stream>>>(nu_log, theta_log, gamma_log,
                                           B_real, B_img, lambda, BnR, BnI);

    dim3 grid(NCHUNK, D_H / 16);
    chunk_scan_kernel<false><<<grid, 32, 0, stream>>>(
        x, BnR, BnI, lambda, nullptr, agg, nullptr);

    combine_kernel<<<(D_H + 255) / 256, 256, 0, stream>>>(lambda, agg, carry);

    chunk_scan_kernel<true><<<grid, 32, 0, stream>>>(
        x, BnR, BnI, lambda, carry, nullptr, (float*)d_out);
}